// InvariantPointAttention_80092550136299
// MI455X (gfx1250) — compile-verified
//
#include <hip/hip_runtime.h>
#include <hip/hip_bf16.h>
#include <math.h>

// -----------------------------------------------------------------------------
// Invariant Point Attention for MI455X (gfx1250), wave32, f32 WMMA everywhere.
// Memory-bound on inputs_2d (2 x 302MB HBM passes); f32 WMMA 16x16x4 keeps
// full precision while lighting up the matrix pipes. All contiguous per-lane
// operand pairs are fetched as explicit 8-byte vector loads.
// -----------------------------------------------------------------------------

#define NRES 768
#define NH   12
#define CIN  384
#define PDIM 128
#define QKW  28      // 16 scalar + 12 point dims, fused QK contraction width
#define VW   40      // 16 scalar V + 24 point V, fused
#define FEATW 2112   // H*(16 + 4*8 + 128)

typedef __attribute__((ext_vector_type(2))) float v2f;
typedef __attribute__((ext_vector_type(8))) float v8f;

__device__ __forceinline__ v8f wmma4(v2f a, v2f b, v8f c) {
    return __builtin_amdgcn_wmma_f32_16x16x4_f32(false, a, false, b, (short)0, c,
                                                 false, false);
}

__device__ __forceinline__ v2f ld2(const float* __restrict__ p) {
    return *(const v2f*)p;   // caller guarantees 8B alignment
}

// ---------------- projections: C[M x Nw] = A[M x K] @ W[K x Nw] + bias -------
__global__ void k_proj(const float* __restrict__ A, const float* __restrict__ W,
                       const float* __restrict__ bias, float* __restrict__ C,
                       int K, int Nw) {
    const int lane = threadIdx.x, half = lane >> 4, l16 = lane & 15;
    const float* arow = A + (size_t)(blockIdx.y * 16 + l16) * K;
    const int nn = blockIdx.x * 16 + l16;   // B col for this lane
    v8f acc = {};
    for (int k0 = 0; k0 < K; k0 += 4) {
        const int ka = k0 + 2 * half;
        const v2f a = ld2(arow + ka);
        v2f b;
        b.x = W[(size_t)ka * Nw + nn];
        b.y = W[(size_t)(ka + 1) * Nw + nn];
        acc = wmma4(a, b, acc);
    }
    const float bb = bias[nn];
    for (int r = 0; r < 8; ++r) {
        const int row = blockIdx.y * 16 + 8 * half + r;
        C[(size_t)row * Nw + nn] = acc[r] + bb;
    }
}

// ---------------- point transforms + fused Q/K/V construction ----------------
__global__ void k_prep(const float* __restrict__ Q, const float* __restrict__ KV,
                       const float* __restrict__ QP, const float* __restrict__ KVP,
                       const float* __restrict__ rot, const float* __restrict__ trans,
                       const float* __restrict__ tpw,
                       float* __restrict__ Qcat, float* __restrict__ Kcat,
                       float* __restrict__ qb, float* __restrict__ kb,
                       float* __restrict__ Vc) {
    int idx = blockIdx.x * blockDim.x + threadIdx.x;
    if (idx >= NRES * NH) return;
    const int i = idx / NH, h = idx % NH;
    const float x0 = tpw[h];
    const float sp = (x0 > 20.f) ? x0 : log1pf(__expf(x0));   // softplus
    const float pw = sqrtf(1.f / 54.f) * sp;                  // sqrt(1/(3*18))
    const float sw = sqrtf(1.f / 48.f);                       // sqrt(1/(3*16))
    float R[9], T[3];
    for (int r = 0; r < 9; ++r) R[r] = rot[r * NRES + i];
    for (int r = 0; r < 3; ++r) T[r] = trans[r * NRES + i];

    float* qrow = Qcat + (size_t)idx * QKW;
    float* krow = Kcat + (size_t)idx * QKW;
    float* vrow = Vc   + (size_t)idx * VW;
    for (int c = 0; c < 16; ++c) {
        qrow[c] = sw * Q[(size_t)i * 192 + h * 16 + c];
        krow[c] = KV[(size_t)i * 384 + h * 32 + c];
        vrow[c] = KV[(size_t)i * 384 + h * 32 + 16 + c];
    }
    // query points (4 per head), rigid transform, fold pw into Q side
    float qn = 0.f;
    for (int p = 0; p < 4; ++p) {
        const float x = QP[(size_t)i * 144 +       h * 4 + p];
        const float y = QP[(size_t)i * 144 +  48 + h * 4 + p];
        const float z = QP[(size_t)i * 144 +  96 + h * 4 + p];
        const float g0 = R[0] * x + R[1] * y + R[2] * z + T[0];
        const float g1 = R[3] * x + R[4] * y + R[5] * z + T[1];
        const float g2 = R[6] * x + R[7] * y + R[8] * z + T[2];
        qrow[16 + 0 + p] = pw * g0;
        qrow[16 + 4 + p] = pw * g1;
        qrow[16 + 8 + p] = pw * g2;
        qn += g0 * g0 + g1 * g1 + g2 * g2;
    }
    qb[idx] = -0.5f * pw * qn;
    // key (4) + value (8) points
    float kn = 0.f;
    for (int s = 0; s < 12; ++s) {
        const float x = KVP[(size_t)i * 432 +        h * 12 + s];
        const float y = KVP[(size_t)i * 432 + 144 + h * 12 + s];
        const float z = KVP[(size_t)i * 432 + 288 + h * 12 + s];
        const float g0 = R[0] * x + R[1] * y + R[2] * z + T[0];
        const float g1 = R[3] * x + R[4] * y + R[5] * z + T[1];
        const float g2 = R[6] * x + R[7] * y + R[8] * z + T[2];
        if (s < 4) {
            krow[16 + 0 + s] = g0;
            krow[16 + 4 + s] = g1;
            krow[16 + 8 + s] = g2;
            kn += g0 * g0 + g1 * g1 + g2 * g2;
        } else {
            const int p = s - 4;
            vrow[16 +  0 + p] = g0;
            vrow[16 +  8 + p] = g1;
            vrow[16 + 16 + p] = g2;
        }
    }
    kb[idx] = -0.5f * pw * kn;
}

// ---------------- logits = Qcat.Kcat + qbias + kbias - mask ------------------
__global__ void k_qk(const float* __restrict__ Qcat, const float* __restrict__ Kcat,
                     const float* __restrict__ qb, const float* __restrict__ kb,
                     const float* __restrict__ mask, float* __restrict__ L) {
    const int jt = blockIdx.x, it = blockIdx.y, h = blockIdx.z;
    const int lane = threadIdx.x, half = lane >> 4, l16 = lane & 15;
    const float* qrow = Qcat + ((size_t)(it * 16 + l16) * NH + h) * QKW; // A row i
    const float* krow = Kcat + ((size_t)(jt * 16 + l16) * NH + h) * QKW; // B col j
    v8f acc = {};
    for (int k0 = 0; k0 < QKW; k0 += 4) {
        const int ka = k0 + 2 * half;
        acc = wmma4(ld2(qrow + ka), ld2(krow + ka), acc);
    }
    const int j = jt * 16 + l16;
    const float mj = mask[j];
    const float kbj = kb[(size_t)j * NH + h];
    for (int r = 0; r < 8; ++r) {
        const int i = it * 16 + 8 * half + r;
        const float v = acc[r] + qb[(size_t)i * NH + h] + kbj
                        - 50.f * (1.f - mask[i] * mj);
        L[((size_t)i * NH + h) * NRES + j] = v;
    }
}

// ---------------- att2d: logits += sqrt(1/3)*(inputs_2d @ w2d + b2d) ---------
__global__ void k_att2d(const float* __restrict__ in2d, const float* __restrict__ w2d,
                        const float* __restrict__ b2d, float* __restrict__ L) {
    const int jt = blockIdx.x, i = blockIdx.y;
    const int lane = threadIdx.x, half = lane >> 4, l16 = lane & 15;
    const float* arow = in2d + ((size_t)i * NRES + jt * 16 + l16) * PDIM; // A row j
    const int hcol = l16;                                                 // B col h
    const bool hv = (hcol < NH);
    v8f acc = {};
    for (int k0 = 0; k0 < PDIM; k0 += 4) {
        const int ka = k0 + 2 * half;
        const v2f a = ld2(arow + ka);
        v2f b;
        b.x = hv ? w2d[(size_t)ka * NH + hcol] : 0.f;
        b.y = hv ? w2d[(size_t)(ka + 1) * NH + hcol] : 0.f;
        acc = wmma4(a, b, acc);
    }
    if (hv) {
        const float s = 0.5773502691896258f; // sqrt(1/3)
        const float bb = b2d[hcol];
        for (int r = 0; r < 8; ++r) {
            const int j = jt * 16 + 8 * half + r;
            L[((size_t)i * NH + hcol) * NRES + j] += s * (acc[r] + bb);
        }
    }
}

// ---------------- softmax over j, one wave per (i,h) row ---------------------
__global__ void k_softmax(float* __restrict__ L) {
    float* p = L + (size_t)blockIdx.x * NRES;
    const int lane = threadIdx.x;
    float mx = -1e30f;
    for (int j = lane; j < NRES; j += 32) mx = fmaxf(mx, p[j]);
    for (int o = 16; o > 0; o >>= 1) mx = fmaxf(mx, __shfl_xor(mx, o, 32));
    float sum = 0.f;
    for (int j = lane; j < NRES; j += 32) {
        const float e = __expf(p[j] - mx);
        p[j] = e;
        sum += e;
    }
    for (int o = 16; o > 0; o >>= 1) sum += __shfl_xor(sum, o, 32);
    const float inv = 1.f / sum;
    for (int j = lane; j < NRES; j += 32) p[j] *= inv;
}

// ---------------- attn @ fused V (scalar 16 + point 24) ----------------------
__global__ void k_attnv(const float* __restrict__ L, const float* __restrict__ Vc,
                        float* __restrict__ ResSV) {
    const int it = blockIdx.x, nt = blockIdx.y, h = blockIdx.z;
    const int lane = threadIdx.x, half = lane >> 4, l16 = lane & 15;
    const float* arow = L + ((size_t)(it * 16 + l16) * NH + h) * NRES; // A row i
    const int ncol = nt * 16 + l16;                                   // B col < 40
    const bool nv = (ncol < VW);
    v8f acc = {};
    for (int k0 = 0; k0 < NRES; k0 += 4) {
        const int ka = k0 + 2 * half;
        const v2f a = ld2(arow + ka);
        v2f b;
        b.x = nv ? Vc[((size_t)ka * NH + h) * VW + ncol] : 0.f;
        b.y = nv ? Vc[((size_t)(ka + 1) * NH + h) * VW + ncol] : 0.f;
        acc = wmma4(a, b, acc);
    }
    if (nv) {
        for (int r = 0; r < 8; ++r) {
            const int i = it * 16 + 8 * half + r;
            ResSV[((size_t)i * NH + h) * VW + ncol] = acc[r];
        }
    }
}

// ---------------- res2d: attn[:,i,:] @ inputs_2d[i] -> feat ------------------
__global__ void k_res2d(const float* __restrict__ L, const float* __restrict__ in2d,
                        float* __restrict__ feat) {
    const int ct = blockIdx.x, i = blockIdx.y;
    const int lane = threadIdx.x, half = lane >> 4, l16 = lane & 15;
    const int hrow = l16;            // A row = h, valid < 12
    const bool hv = (hrow < NH);
    const int ccol = ct * 16 + l16;  // B col = channel
    const float* arow = L + ((size_t)i * NH + (hv ? hrow : 0)) * NRES;
    const float* b2 = in2d + (size_t)i * NRES * PDIM;
    const v2f zero2 = {0.f, 0.f};
    v8f acc = {};
    for (int k0 = 0; k0 < NRES; k0 += 4) {
        const int ka = k0 + 2 * half;
        const v2f a = hv ? ld2(arow + ka) : zero2;
        v2f b;
        b.x = b2[(size_t)ka * PDIM + ccol];
        b.y = b2[(size_t)(ka + 1) * PDIM + ccol];
        acc = wmma4(a, b, acc);
    }
    for (int r = 0; r < 8; ++r) {
        const int h = 8 * half + r;
        if (h < NH)
            feat[(size_t)i * FEATW + 576 + h * PDIM + ccol] = acc[r];
    }
}

// ---------------- inverse rigid transform + feature assembly -----------------
__global__ void k_final(const float* __restrict__ ResSV, const float* __restrict__ rot,
                        const float* __restrict__ trans, float* __restrict__ feat) {
    int idx = blockIdx.x * blockDim.x + threadIdx.x;
    if (idx >= NRES * NH) return;
    const int i = idx / NH, h = idx % NH;
    const float* rv = ResSV + (size_t)idx * VW;
    float* f = feat + (size_t)i * FEATW;
    for (int c = 0; c < 16; ++c) f[h * 16 + c] = rv[c];
    float R[9], T[3];
    for (int r = 0; r < 9; ++r) R[r] = rot[r * NRES + i];
    for (int r = 0; r < 3; ++r) T[r] = trans[r * NRES + i];
    for (int p = 0; p < 8; ++p) {
        const float gx = rv[16 + p] - T[0];
        const float gy = rv[24 + p] - T[1];
        const float gz = rv[32 + p] - T[2];
        const float lx = R[0] * gx + R[3] * gy + R[6] * gz;
        const float ly = R[1] * gx + R[4] * gy + R[7] * gz;
        const float lz = R[2] * gx + R[5] * gy + R[8] * gz;
        const int hp = h * 8 + p;
        f[192 + hp] = lx;
        f[288 + hp] = ly;
        f[384 + hp] = lz;
        f[480 + hp] = sqrtf(1e-8f + lx * lx + ly * ly + lz * lz);
    }
}

// ---------------- out = feat @ wout + bout -----------------------------------
__global__ void k_out(const float* __restrict__ feat, const float* __restrict__ wout,
                      const float* __restrict__ bout, float* __restrict__ out) {
    const int ct = blockIdx.x, it = blockIdx.y;
    const int lane = threadIdx.x, half = lane >> 4, l16 = lane & 15;
    const float* arow = feat + (size_t)(it * 16 + l16) * FEATW;
    const int nn = ct * 16 + l16;
    v8f acc = {};
    for (int k0 = 0; k0 < FEATW; k0 += 4) {
        const int ka = k0 + 2 * half;
        const v2f a = ld2(arow + ka);
        v2f b;
        b.x = wout[(size_t)ka * CIN + nn];
        b.y = wout[(size_t)(ka + 1) * CIN + nn];
        acc = wmma4(a, b, acc);
    }
    const float bb = bout[nn];
    for (int r = 0; r < 8; ++r) {
        const int row = it * 16 + 8 * half + r;
        out[(size_t)row * CIN + nn] = acc[r] + bb;
    }
}

// -----------------------------------------------------------------------------
extern "C" void kernel_launch(void* const* d_in, const int* in_sizes, int n_in,
                              void* d_out, int out_size, void* d_ws, size_t ws_size,
                              hipStream_t stream) {
    const float* in1d  = (const float*)d_in[0];
    const float* in2d  = (const float*)d_in[1];
    const float* mask  = (const float*)d_in[2];
    const float* rotn  = (const float*)d_in[3];
    const float* trans = (const float*)d_in[4];
    const float* wq    = (const float*)d_in[5];
    const float* bq    = (const float*)d_in[6];
    const float* wkv   = (const float*)d_in[7];
    const float* bkv   = (const float*)d_in[8];
    const float* wqp   = (const float*)d_in[9];
    const float* bqp   = (const float*)d_in[10];
    const float* wkvp  = (const float*)d_in[11];
    const float* bkvp  = (const float*)d_in[12];
    const float* tpw   = (const float*)d_in[13];
    const float* w2d   = (const float*)d_in[14];
    const float* b2d   = (const float*)d_in[15];
    const float* wout  = (const float*)d_in[16];
    const float* bout  = (const float*)d_in[17];
    float* out = (float*)d_out;

    float* ws = (float*)d_ws;
    size_t off = 0;
    float* Qp   = ws + off; off += (size_t)NRES * 192;       // q proj
    float* KVp  = ws + off; off += (size_t)NRES * 384;       // kv proj
    float* QPp  = ws + off; off += (size_t)NRES * 144;       // q point proj
    float* KVPp = ws + off; off += (size_t)NRES * 432;       // kv point proj
    float* Qcat = ws + off; off += (size_t)NRES * NH * QKW;
    float* Kcat = ws + off; off += (size_t)NRES * NH * QKW;
    float* qb   = ws + off; off += (size_t)NRES * NH;
    float* kb   = ws + off; off += (size_t)NRES * NH;
    float* Vc   = ws + off; off += (size_t)NRES * NH * VW;
    float* RSV  = ws + off; off += (size_t)NRES * NH * VW;
    float* feat = ws + off; off += (size_t)NRES * FEATW;
    float* L    = ws + off; off += (size_t)NH * NRES * NRES; // ~28 MB; total ~44 MB

    const dim3 wave(32);
    // projections
    k_proj<<<dim3(192 / 16, NRES / 16), wave, 0, stream>>>(in1d, wq,   bq,   Qp,   CIN, 192);
    k_proj<<<dim3(384 / 16, NRES / 16), wave, 0, stream>>>(in1d, wkv,  bkv,  KVp,  CIN, 384);
    k_proj<<<dim3(144 / 16, NRES / 16), wave, 0, stream>>>(in1d, wqp,  bqp,  QPp,  CIN, 144);
    k_proj<<<dim3(432 / 16, NRES / 16), wave, 0, stream>>>(in1d, wkvp, bkvp, KVPp, CIN, 432);
    // fused Q/K/V construction with rigid transforms
    k_prep<<<dim3((NRES * NH + 255) / 256), dim3(256), 0, stream>>>(
        Qp, KVp, QPp, KVPp, rotn, trans, tpw, Qcat, Kcat, qb, kb, Vc);
    // logits
    k_qk<<<dim3(NRES / 16, NRES / 16, NH), wave, 0, stream>>>(Qcat, Kcat, qb, kb, mask, L);
    k_att2d<<<dim3(NRES / 16, NRES), wave, 0, stream>>>(in2d, w2d, b2d, L);
    // softmax
    k_softmax<<<dim3(NRES * NH), wave, 0, stream>>>(L);
    // attention outputs
    k_attnv<<<dim3(NRES / 16, 3, NH), wave, 0, stream>>>(L, Vc, RSV);
    k_res2d<<<dim3(PDIM / 16, NRES), wave, 0, stream>>>(L, in2d, feat);
    k_final<<<dim3((NRES * NH + 255) / 256), dim3(256), 0, stream>>>(RSV, rotn, trans, feat);
    // output projection
    k_out<<<dim3(CIN / 16, NRES / 16), wave, 0, stream>>>(feat, wout, bout, out);
}